// YOLOLoss_75909251989904
// MI455X (gfx1250) — compile-verified
//
#include <hip/hip_runtime.h>
#include <math.h>

typedef float v2f __attribute__((ext_vector_type(2)));
typedef float v8f __attribute__((ext_vector_type(8)));

#define EPSF 1e-7f

// ---- Workspace layout (floats) ----
#define OFF_T0     0        // tobj layer0: 16*80*80 = 102400
#define OFF_T1     102400   // tobj layer1: 16*40*40 = 25600
#define OFF_T2     128000   // tobj layer2: 16*20*20 = 6400
#define OFF_EPART  134400   // 36 entry-blocks x {lbox, lcls, n} = 108
#define OFF_OBJ    134512   // 394 per-block lobj partials
#define WS_FLOATS  134912

// k_obj block decomposition: 1024 cells / block
// layer cells: 307200, 76800, 19200 -> blocks: 300, 75, 19 (total 394)
#define OBJ_NB0 300
#define OBJ_NB1 75
#define OBJ_NB2 19
#define OBJ_TOTAL (OBJ_NB0 + OBJ_NB1 + OBJ_NB2)

__device__ __constant__ float d_ANC[3][3][2] = {
  {{1.25f, 1.625f},  {2.0f, 3.75f},     {4.125f, 2.875f}},
  {{1.875f, 3.8125f},{3.875f, 2.8125f}, {3.6875f, 7.4375f}},
  {{3.625f, 2.8125f},{4.875f, 6.1875f}, {11.65625f, 10.1875f}}
};
__device__ __constant__ float d_OFFX[5] = {0.f, 0.5f, 0.f, -0.5f, 0.f};
__device__ __constant__ float d_OFFY[5] = {0.f, 0.f, 0.5f, 0.f, -0.5f};
__device__ __constant__ int   d_H[5]    = {80, 40, 20, 0, 0};

// ---- CDNA5 wave32 sum via V_WMMA_F32_16X16X4_F32 (A = ones -> column sums).
// B's 4x16 layout maps lanes n and n+16 to the same column, so one WMMA
// pair-sums across the lane halves; xor-shuffles finish the 16 columns.
__device__ __forceinline__ float wave_sum(float v) {
  v2f a; a[0] = 1.0f; a[1] = 1.0f;   // A[m,k] = 1 for all m,k
  v2f b; b[0] = v;    b[1] = 0.0f;   // one B slot per lane holds the partial
  v8f c = {};
  c = __builtin_amdgcn_wmma_f32_16x16x4_f32(false, a, false, b, (short)0, c,
                                            false, false);
  float p = c[0];                    // = v[lane%16] + v[lane%16 + 16]
  p += __shfl_xor(p, 1);
  p += __shfl_xor(p, 2);
  p += __shfl_xor(p, 4);
  p += __shfl_xor(p, 8);
  return p;                          // full 32-lane sum in every lane
}

// Block-wide sum; result valid in thread 0. Must be called in uniform flow
// with a full block (EXEC all-ones at the WMMA).
__device__ __forceinline__ float block_sum(float v, float* sm) {
  float w = wave_sum(v);
  int wid = threadIdx.x >> 5, lane = threadIdx.x & 31;
  if (lane == 0) sm[wid] = w;
  __syncthreads();
  float r = 0.f;
  if (threadIdx.x == 0) {
    int nw = blockDim.x >> 5;
    for (int k = 0; k < nw; ++k) r += sm[k];
  }
  __syncthreads();
  return r;
}

__device__ __forceinline__ float softplus_f(float x) {
  return fmaxf(x, 0.f) + log1pf(expf(-fabsf(x)));
}
__device__ __forceinline__ float sigm(float x) { return 1.0f / (1.0f + expf(-x)); }

__device__ __forceinline__ float ciou_fn(float px, float py, float pw, float ph,
                                         float tx, float ty, float tw, float th) {
  float b1x1 = px - pw * 0.5f, b1x2 = px + pw * 0.5f;
  float b1y1 = py - ph * 0.5f, b1y2 = py + ph * 0.5f;
  float b2x1 = tx - tw * 0.5f, b2x2 = tx + tw * 0.5f;
  float b2y1 = ty - th * 0.5f, b2y2 = ty + th * 0.5f;
  float iw = fmaxf(fminf(b1x2, b2x2) - fmaxf(b1x1, b2x1), 0.f);
  float ih = fmaxf(fminf(b1y2, b2y2) - fmaxf(b1y1, b2y1), 0.f);
  float inter = iw * ih;
  float w1 = b1x2 - b1x1, h1 = b1y2 - b1y1 + EPSF;
  float w2 = b2x2 - b2x1, h2 = b2y2 - b2y1 + EPSF;
  float uni = w1 * h1 + w2 * h2 - inter + EPSF;
  float iou = inter / uni;
  float cw = fmaxf(b1x2, b2x2) - fminf(b1x1, b2x1);
  float ch = fmaxf(b1y2, b2y2) - fminf(b1y1, b2y1);
  float c2 = cw * cw + ch * ch + EPSF;
  float dx = b2x1 + b2x2 - b1x1 - b1x2;
  float dy = b2y1 + b2y2 - b1y1 - b1y2;
  float rho2 = (dx * dx + dy * dy) * 0.25f;
  float dv = atanf(w2 / h2) - atanf(w1 / h1);
  float v = (4.0f / 9.869604401089358f) * dv * dv;  // 4/pi^2
  float alpha = v / (v - iou + 1.0f + EPSF);
  return iou - (rho2 / c2 + v * alpha);
}

__global__ void k_zero(float* __restrict__ ws, int n) {
  int i = blockIdx.x * blockDim.x + threadIdx.x;
  if (i < n) ws[i] = 0.f;
}

// 12 blocks x 256 threads per layer: one target candidate per thread.
// Produces per-block {lbox,lcls,n} partials + exact tobj scatter (uint atomicMax).
__global__ void k_entries(const float* __restrict__ p0, const float* __restrict__ p1,
                          const float* __restrict__ p2, const float* __restrict__ tgt,
                          float* __restrict__ ws) {
  int l   = blockIdx.x / 12;
  int sub = blockIdx.x % 12;
  int h = d_H[l], w = h;
  const float* pred = (l == 0) ? p0 : (l == 1) ? p1 : p2;
  float* tobj = ws + ((l == 0) ? OFF_T0 : (l == 1) ? OFF_T1 : OFF_T2);
  float aw0 = d_ANC[l][0][0], ah0 = d_ANC[l][0][1];  // pwh uses anchors[0]

  float lb = 0.f, lc = 0.f, nn = 0.f;
  int e = sub * 256 + (int)threadIdx.x;
  if (e < 3000) {
    int o = e / 600, rem = e % 600, a = rem / 200, i = rem % 200;
    const float* t = tgt + i * 6;
    float gx = t[2] * (float)w, gy = t[3] * (float)h;
    float gw = t[4] * (float)w, gh = t[5] * (float)h;
    float rw = gw / d_ANC[l][a][0], rh = gh / d_ANC[l][a][1];
    float mr = fmaxf(fmaxf(rw, 1.f / rw), fmaxf(rh, 1.f / rh));
    bool m0 = mr < 4.0f;
    bool sel;
    if (o == 0)      sel = true;
    else if (o == 1) sel = ((gx - floorf(gx)) < 0.5f) && (gx > 1.0f);
    else if (o == 2) sel = ((gy - floorf(gy)) < 0.5f) && (gy > 1.0f);
    else if (o == 3) { float gxi = (float)w - gx; sel = ((gxi - floorf(gxi)) < 0.5f) && (gxi > 1.0f); }
    else             { float gyi = (float)h - gy; sel = ((gyi - floorf(gyi)) < 0.5f) && (gyi > 1.0f); }
    bool keep = m0 && sel;
    float fm = keep ? 1.f : 0.f;

    int gi = (int)floorf(gx - d_OFFX[o]); gi = min(max(gi, 0), w - 1);
    int gj = (int)floorf(gy - d_OFFY[o]); gj = min(max(gj, 0), h - 1);
    int bi = (int)t[0];
    int ci = (int)t[1];
    // reference uses a = ones_like(b) -> anchor channel 1 for gather & scatter
    size_t base = ((size_t)((bi * 3 + 1) * h + gj) * w + gi) * 85;

    float s0 = pred[base + 0], s1 = pred[base + 1];
    float s2 = pred[base + 2], s3 = pred[base + 3];
    float pxc = sigm(s0) * 2.f - 0.5f;
    float pyc = sigm(s1) * 2.f - 0.5f;
    float pwv = sigm(s2) * 2.f; pwv = pwv * pwv * aw0;
    float phv = sigm(s3) * 2.f; phv = phv * phv * ah0;
    float tbx = gx - (float)gi, tby = gy - (float)gj;
    float iou = ciou_fn(pxc, pyc, pwv, phv, tbx, tby, gw, gh);
    lb = (1.f - iou) * fm;

    // sum_j bce(x_j, onehot_j) = sum_j softplus(x_j) - x_class
    float cls = 0.f;
    for (int j = 0; j < 80; ++j) cls += softplus_f(pred[base + 5 + j]);
    cls -= pred[base + 5 + ci];
    lc = cls * fm;
    nn = fm;

    if (keep) {
      float tv = fmaxf(iou, 0.f);  // non-negative -> uint-bit atomicMax is exact
      atomicMax((unsigned int*)&tobj[(size_t)(bi * h + gj) * w + gi],
                __float_as_uint(tv));
    }
  }

  __shared__ float sm[8];
  float rb = block_sum(lb, sm);
  float rc = block_sum(lc, sm);
  float rn = block_sum(nn, sm);
  if (threadIdx.x == 0) {
    float* q = ws + OFF_EPART + (size_t)blockIdx.x * 3;
    q[0] = rb; q[1] = rc; q[2] = rn;
  }
}

// Streaming lobj reduction: 1024 cells per 256-thread block (394 blocks total)
// to generate enough outstanding loads to saturate HBM. Deterministic partials.
__global__ void k_obj(const float* __restrict__ p0, const float* __restrict__ p1,
                      const float* __restrict__ p2, float* __restrict__ ws) {
  int blk = blockIdx.x;
  int l, bloc;
  if (blk < OBJ_NB0)                { l = 0; bloc = blk; }
  else if (blk < OBJ_NB0 + OBJ_NB1) { l = 1; bloc = blk - OBJ_NB0; }
  else                              { l = 2; bloc = blk - OBJ_NB0 - OBJ_NB1; }
  int h = d_H[l], hw = h * h;
  int cells = 48 * hw;  // 16 * 3 * h * w
  const float* pred = (l == 0) ? p0 : (l == 1) ? p1 : p2;
  const float* tobj = ws + ((l == 0) ? OFF_T0 : (l == 1) ? OFF_T1 : OFF_T2);

  float acc = 0.f;
  int base = bloc * 1024;
  #pragma unroll
  for (int it = 0; it < 4; ++it) {
    int cell = base + it * 256 + (int)threadIdx.x;
    if (cell < cells) {
      int p = cell % hw;
      int ba = cell / hw;
      int a = ba % 3, b = ba / 3;
      float x = __builtin_nontemporal_load(&pred[(size_t)cell * 85 + 4]);
      float tv = (a == 1) ? tobj[b * hw + p] : 0.f;
      acc += fmaxf(x, 0.f) - x * tv + log1pf(expf(-fabsf(x)));
    }
  }
  __shared__ float sm[8];
  float r = block_sum(acc, sm);
  if (threadIdx.x == 0) ws[OFF_OBJ + blk] = r;
}

// Single 256-thread block: reduce all partials (parallel, deterministic) and
// combine with gains.
__global__ void k_final(const float* __restrict__ ws, float* __restrict__ out) {
  __shared__ float sm[8];
  __shared__ float res[12];
  const int ostart[4] = {0, OBJ_NB0, OBJ_NB0 + OBJ_NB1, OBJ_TOTAL};

  // per-layer lobj partial sums
  for (int l = 0; l < 3; ++l) {
    float v = 0.f;
    int cnt = ostart[l + 1] - ostart[l];
    for (int k = (int)threadIdx.x; k < cnt; k += 256)
      v += ws[OFF_OBJ + ostart[l] + k];
    float r = block_sum(v, sm);
    if (threadIdx.x == 0) res[l] = r;
    __syncthreads();
  }
  // per-layer entry partial sums (12 blocks each)
  for (int l = 0; l < 3; ++l) {
    float vb = 0.f, vc = 0.f, vn = 0.f;
    if (threadIdx.x < 12) {
      const float* q = ws + OFF_EPART + (size_t)(l * 12 + threadIdx.x) * 3;
      vb = q[0]; vc = q[1]; vn = q[2];
    }
    float rb = block_sum(vb, sm);
    float rc = block_sum(vc, sm);
    float rn = block_sum(vn, sm);
    if (threadIdx.x == 0) {
      res[3 + l * 3] = rb; res[4 + l * 3] = rc; res[5 + l * 3] = rn;
    }
    __syncthreads();
  }

  if (threadIdx.x == 0) {
    const int hh[3] = {80, 40, 20};
    float lbox = 0.f, lcls = 0.f, lobj = 0.f;
    for (int l = 0; l < 3; ++l) {
      float n = res[5 + l * 3];
      lbox += res[3 + l * 3] / n;
      lcls += res[4 + l * 3] / (n * 80.f);
      lobj += res[l] / (float)(48 * hh[l] * hh[l]);
    }
    lbox *= 0.05f;  // BOX_GAIN
    lcls *= 0.5f;   // CLS_GAIN
    float loss = lbox + lobj + lcls;
    out[0] = loss;
    out[1] = lbox;
    out[2] = lobj;
    out[3] = lcls;
    out[4] = loss;
  }
}

extern "C" void kernel_launch(void* const* d_in, const int* in_sizes, int n_in,
                              void* d_out, int out_size, void* d_ws, size_t ws_size,
                              hipStream_t stream) {
  const float* p0 = (const float*)d_in[0];
  const float* p1 = (const float*)d_in[1];
  const float* p2 = (const float*)d_in[2];
  const float* tg = (const float*)d_in[3];
  float* ws  = (float*)d_ws;
  float* out = (float*)d_out;

  k_zero<<<(WS_FLOATS + 255) / 256, 256, 0, stream>>>(ws, WS_FLOATS);
  k_entries<<<36, 256, 0, stream>>>(p0, p1, p2, tg, ws);
  k_obj<<<OBJ_TOTAL, 256, 0, stream>>>(p0, p1, p2, ws);
  k_final<<<1, 256, 0, stream>>>(ws, out);
}